// DASAndPixelInterpolator_MSOT_65901978189952
// MI455X (gfx1250) — compile-verified
//
#include <hip/hip_runtime.h>

// Problem constants (from the reference)
#define ROI    128
#define NS     64          // sensors (LV_END - LV_START)
#define TSAMP  2030
#define NBATCH 64
#define PIX    (ROI * ROI) // 16384
#define LSTRIDE 129        // LDS row stride in elements (bank-conflict padding)
#define BPB    8           // batches per block

typedef __attribute__((ext_vector_type(2))) float v2f;
typedef __attribute__((ext_vector_type(8))) float v8f;

#if __has_builtin(__builtin_amdgcn_wmma_f32_16x16x4_f32)
#define HAVE_WMMA_F32X4 1
#else
#define HAVE_WMMA_F32X4 0
#endif

__global__ void das_init_ws(unsigned* __restrict__ wsmax) {
    if (threadIdx.x < NBATCH) wsmax[threadIdx.x] = 0u;  // relu'd data => max >= 0, bits monotonic
}

__global__ __launch_bounds__(256)
void das_pass1(const float* __restrict__ sino,      // (B, 1, NS, TSAMP)
               const float* __restrict__ weights,   // (ROI, ROI, NS)
               const int*   __restrict__ tidx,      // (ROI, ROI, NS)
               float*       __restrict__ das_out,   // (B, 1, ROI, ROI), unnormalized relu'd
               float*       __restrict__ pix_out,   // (B, NS, ROI, ROI)
               unsigned*    __restrict__ wsmax)     // 64 per-batch maxima (float bits)
{
    __shared__ unsigned short lds_t[NS * LSTRIDE];   // clamped time index (< 2030 fits u16)
    __shared__ float          lds_w[NS * LSTRIDE];   // validity-masked weight
    __shared__ float          lds_das[8 * 16];       // per-wave DAS staging
    __shared__ float          lds_red[8];            // per-wave max staging

    const int tid = threadIdx.x;
    const int y   = blockIdx.x;              // image row
    const int b0  = blockIdx.y * BPB;        // batch chunk

    // ---- Cooperative load of row geometry, transposed [x][s] -> [s][x] ----
    const int rowbase = y * ROI * NS;
    for (int j = tid; j < ROI * NS; j += 256) {
        const int x = j >> 6;                // 64 sensors per pixel
        const int s = j & (NS - 1);
        const int t = tidx[rowbase + j];
        const float w = weights[rowbase + j];
        const bool valid = (t >= 0) && (t < TSAMP);
        int tc = t < 0 ? 0 : (t >= TSAMP ? TSAMP - 1 : t);
        lds_t[s * LSTRIDE + x] = (unsigned short)tc;
        lds_w[s * LSTRIDE + x] = valid ? w : 0.0f;
    }
    __syncthreads();

    const int wave = tid >> 5;
    const int lane = tid & 31;
    const int half = lane >> 4;              // WMMA A-matrix K-pair selector
    const int x    = (wave << 4) + (lane & 15);   // wave owns 16 consecutive pixels

    const v2f bones = {1.0f, 1.0f};          // all-ones B matrix -> row-sum reduction

    for (int bi = 0; bi < BPB; ++bi) {
        const int b = b0 + bi;
        const float* __restrict__ sb = sino + (size_t)b * NS * TSAMP;
        float* __restrict__ pb = pix_out + (size_t)b * NS * PIX + (size_t)y * ROI + x;

        v8f acc = {0.f, 0.f, 0.f, 0.f, 0.f, 0.f, 0.f, 0.f};
#if !HAVE_WMMA_F32X4
        float ssum = 0.0f;
#endif
        #pragma unroll 4
        for (int s0 = 0; s0 < NS; s0 += 4) {
            const int sA = s0 + (half << 1);            // lane's two sensors in this K=4 chunk
            const int t0 = (int)lds_t[ sA      * LSTRIDE + x];
            const int t1 = (int)lds_t[(sA + 1) * LSTRIDE + x];
            const float w0 = lds_w[ sA      * LSTRIDE + x];
            const float w1 = lds_w[(sA + 1) * LSTRIDE + x];
            const float a0 = w0 * sb[(size_t)sA * TSAMP + t0];        // L2-resident gather
            const float a1 = w1 * sb[(size_t)(sA + 1) * TSAMP + t1];
            pb[(size_t) sA      * PIX] = a0;            // two 64B contiguous segments / store
            pb[(size_t)(sA + 1) * PIX] = a1;
            v2f av = {a0, a1};                          // A-matrix: M=lane&15, K=2*half+{0,1}
#if HAVE_WMMA_F32X4
            acc = __builtin_amdgcn_wmma_f32_16x16x4_f32(
                false, av, false, bones, (short)0, acc, false, false);
#else
            ssum += a0 + a1;
            (void)acc;
#endif
        }

        __syncthreads();   // protect staging (WAR vs previous batch iteration)
#if HAVE_WMMA_F32X4
        // D[m][n] identical for all n; lanes with N==0 (lane 0 / 16) hold M=r / M=8+r in acc[r]
        if ((lane & 15) == 0) {
            #pragma unroll
            for (int r = 0; r < 8; ++r)
                lds_das[(wave << 4) + (half << 3) + r] = acc[r];
        }
#endif
        __syncthreads();
#if HAVE_WMMA_F32X4
        float dval = lds_das[(wave << 4) + (lane & 15)];
#else
        float dval = ssum + __shfl_xor(ssum, 16, 32);   // lane pair (l, l+16) shares a pixel
#endif
        dval = fmaxf(dval, 0.0f);                       // relu before per-batch max (reference order)
        if (lane < 16)
            das_out[(size_t)b * PIX + y * ROI + x] = dval;

        // block max -> one atomic per (block, batch)
        float m = dval;
        #pragma unroll
        for (int off = 16; off > 0; off >>= 1)
            m = fmaxf(m, __shfl_xor(m, off, 32));
        if (lane == 0) lds_red[wave] = m;
        __syncthreads();
        if (tid == 0) {
            float bm = lds_red[0];
            #pragma unroll
            for (int w2 = 1; w2 < 8; ++w2) bm = fmaxf(bm, lds_red[w2]);
            atomicMax(&wsmax[b], __float_as_uint(bm));
        }
    }
}

__global__ __launch_bounds__(256)
void das_norm(float* __restrict__ das, const unsigned* __restrict__ wsmax) {
    const int b = blockIdx.y;
    const int p = blockIdx.x * 256 + threadIdx.x;
    const float m = __uint_as_float(wsmax[b]);
    const float mm = (m > 1e-8f) ? m : 1.0f;
    das[(size_t)b * PIX + p] = das[(size_t)b * PIX + p] / mm;
}

extern "C" void kernel_launch(void* const* d_in, const int* in_sizes, int n_in,
                              void* d_out, int out_size, void* d_ws, size_t ws_size,
                              hipStream_t stream) {
    (void)in_sizes; (void)n_in; (void)out_size; (void)ws_size;
    const float* sino    = (const float*)d_in[0];   // (64,1,64,2030) f32
    const float* weights = (const float*)d_in[1];   // (128,128,64)   f32
    const int*   tidx    = (const int*)  d_in[2];   // (128,128,64)   i32
    // d_in[3] (valid_mask) intentionally unused: recomputed from tidx.

    float* das = (float*)d_out;                       // first output, 64*16384 floats
    float* pix = das + (size_t)NBATCH * PIX;          // second output, 64*64*16384 floats
    unsigned* wsmax = (unsigned*)d_ws;

    hipLaunchKernelGGL(das_init_ws, dim3(1), dim3(64), 0, stream, wsmax);
    hipLaunchKernelGGL(das_pass1,
                       dim3(ROI, NBATCH / BPB), dim3(256), 0, stream,
                       sino, weights, tidx, das, pix, wsmax);
    hipLaunchKernelGGL(das_norm,
                       dim3(PIX / 256, NBATCH), dim3(256), 0, stream,
                       das, wsmax);
}